// PersistenceHomologyTracker_9010841387126
// MI455X (gfx1250) — compile-verified
//
#include <hip/hip_runtime.h>
#include <hip/hip_bf16.h>
#include <stdint.h>

typedef float v2f __attribute__((ext_vector_type(2)));
typedef float v8f __attribute__((ext_vector_type(8)));

#define TT 4096
#define CC 256
#define NMAT 8   // 2 inputs * 4 batches

// ---------------------------------------------------------------------------
// Kernel S: per-row sum of squares, sq[sel*16384 + b*4096 + t]; also zero the
// edge counters so every launch is deterministic.
// ---------------------------------------------------------------------------
__global__ void sq_kernel(const float* __restrict__ xb, const float* __restrict__ xa,
                          float* __restrict__ sq, int* __restrict__ edgeCnt) {
  const int wave = threadIdx.x >> 5;
  const int lane = threadIdx.x & 31;
  const int row  = blockIdx.x * 8 + wave;          // 0 .. 32767
  const int sel  = row >> 14;                       // 16384 rows per input
  const int r    = row & 16383;
  const float* X = sel ? xa : xb;
  const float* p = X + (size_t)r * CC;
  float s = 0.f;
  #pragma unroll
  for (int t = 0; t < CC; t += 32) { float v = p[t + lane]; s += v * v; }
  for (int off = 16; off > 0; off >>= 1) s += __shfl_xor(s, off, 32);
  if (lane == 0) sq[row] = s;
  if (blockIdx.x == 0 && threadIdx.x < NMAT) edgeCnt[threadIdx.x] = 0;
}

// ---------------------------------------------------------------------------
// Kernel A: WMMA f32 Gram tiles -> d2 threshold -> packed bit adjacency +
// edge popcounts. One wave per 16x16 tile; block = 4 waves sharing the i-strip.
// ---------------------------------------------------------------------------
__global__ __launch_bounds__(128)
void adj_kernel(const float* __restrict__ xb, const float* __restrict__ xa,
                const float* __restrict__ sq, uint16_t* __restrict__ adj,
                int* __restrict__ edgeCnt) {
  const int sb   = blockIdx.z;                      // 0..7  (sel*4 + batch)
  const int sel  = sb >> 2, bb = sb & 3;
  const int wave = threadIdx.x >> 5;
  const int lane = threadIdx.x & 31;
  const int i0   = blockIdx.y * 16;
  const int j0   = blockIdx.x * 64 + wave * 16;

  const float* X = (sel ? xa : xb) + (size_t)bb * TT * CC;
  const int half = lane >> 4;                       // 0: K=0,1   1: K=2,3
  const int lrow = lane & 15;
  const int koff = half * 2;
  const float* ai = X + (size_t)(i0 + lrow) * CC + koff;  // A: rows i0..i0+15
  const float* bj = X + (size_t)(j0 + lrow) * CC + koff;  // B: cols j0..j0+15

  v8f acc = {};
  #pragma unroll 8
  for (int k0 = 0; k0 < CC; k0 += 4) {
    v2f a = *(const v2f*)(ai + k0);
    v2f b = *(const v2f*)(bj + k0);
    acc = __builtin_amdgcn_wmma_f32_16x16x4_f32(
        /*neg_a=*/false, a, /*neg_b=*/false, b,
        /*c_mod=*/(short)0, acc, /*reuse_a=*/false, /*reuse_b=*/false);
  }

  const float eps2 = (float)(22.6 * 22.6);          // matches jnp f32 compare
  const float* sqb = sq + (size_t)sb * TT;
  const float sqj  = sqb[j0 + lrow];
  uint32_t cnt = 0;
  uint16_t* rowbase = adj + (size_t)sb * TT * 256 + (j0 >> 4);

  #pragma unroll
  for (int r = 0; r < 8; ++r) {
    // C/D layout: lanes 0-15 -> (row i0+r, col j0+lane); lanes 16-31 -> row i0+r+8
    float sqi = sqb[i0 + r + half * 8];
    float d2  = sqi + sqj - 2.0f * acc[r];
    uint32_t mask = (uint32_t)__ballot(d2 < eps2);  // wave32: bits [31:0]
    cnt += __popc(mask);
    if (lane == r)      rowbase[(size_t)(i0 + r)     * 256] = (uint16_t)(mask & 0xFFFFu);
    if (lane == r + 16) rowbase[(size_t)(i0 + r + 8) * 256] = (uint16_t)(mask >> 16);
  }
  if (lane == 0) atomicAdd(&edgeCnt[sb], (int)cnt);
}

// ---------------------------------------------------------------------------
// Kernel B: connected components (min-label propagation + pointer jumping),
// labels held in LDS; one 1024-thread workgroup per (sel,batch).
// ---------------------------------------------------------------------------
__global__ __launch_bounds__(1024)
void cc_kernel(const uint32_t* __restrict__ adj, const int* __restrict__ edgeCnt,
               int* __restrict__ betti) {
  __shared__ int labels[TT];
  __shared__ int newl[TT];
  __shared__ int sChanged;
  __shared__ int sComp;

  const int sb  = blockIdx.x;
  const uint32_t* A = adj + (size_t)sb * TT * 128;  // 128 words per row
  const int tid = threadIdx.x;
  const int wave = tid >> 5, lane = tid & 31;

  for (int i = tid; i < TT; i += 1024) labels[i] = i;
  __syncthreads();

  for (;;) {
    if (tid == 0) sChanged = 0;
    __syncthreads();

    // prop[row] = min label over neighbors (diag bit includes self)
    for (int row = wave; row < TT; row += 32) {
      const uint32_t* w = A + (size_t)row * 128;
      int m = TT;
      for (int ww = lane; ww < 128; ww += 32) {
        uint32_t bits = w[ww];
        int base = ww << 5;
        while (bits) {
          int b = __builtin_ctz(bits);
          bits &= bits - 1;
          int lab = labels[base + b];
          m = (lab < m) ? lab : m;
        }
      }
      for (int off = 16; off > 0; off >>= 1) {
        int o = __shfl_xor(m, off, 32);
        m = (o < m) ? o : m;
      }
      if (lane == 0) {
        int cur = labels[row];
        newl[row] = (cur < m) ? cur : m;
      }
    }
    __syncthreads();

    // pointer jump: labels[i] = newl[newl[i]]
    int ch = 0;
    for (int i = tid; i < TT; i += 1024) {
      int v = newl[newl[i]];
      if (v != labels[i]) ch = 1;
      labels[i] = v;
    }
    if (ch) sChanged = 1;
    __syncthreads();
    int done = (sChanged == 0);
    __syncthreads();
    if (done) break;
  }

  if (tid == 0) sComp = 0;
  __syncthreads();
  int local = 0;
  for (int i = tid; i < TT; i += 1024) local += (labels[i] == i) ? 1 : 0;
  atomicAdd(&sComp, local);
  __syncthreads();

  if (tid == 0) {
    int nedges = edgeCnt[sb] >> 1;     // sum(adj)//2 (sum is non-negative)
    int b1 = nedges - TT + sComp;
    betti[sb] = b1 > 0 ? b1 : 0;
  }
}

// ---------------------------------------------------------------------------
// Kernel C: mean over batch of (betti_after - betti_before)^2
// ---------------------------------------------------------------------------
__global__ void final_kernel(const int* __restrict__ betti, float* __restrict__ out) {
  if (threadIdx.x == 0 && blockIdx.x == 0) {
    float acc = 0.f;
    for (int b = 0; b < 4; ++b) {
      float d = (float)betti[4 + b] - (float)betti[b];
      acc += d * d;
    }
    out[0] = acc * 0.25f;
  }
}

extern "C" void kernel_launch(void* const* d_in, const int* in_sizes, int n_in,
                              void* d_out, int out_size, void* d_ws, size_t ws_size,
                              hipStream_t stream) {
  const float* xb = (const float*)d_in[0];   // x_before (4,4096,256) f32
  const float* xa = (const float*)d_in[1];   // x_after
  char* ws = (char*)d_ws;

  // workspace layout
  uint16_t* adj = (uint16_t*)ws;                                   // 16 MB packed bits
  float* sq     = (float*)(ws + (size_t)16 * 1024 * 1024);          // 128 KB
  int* edgeCnt  = (int*)(ws + (size_t)16 * 1024 * 1024 + 131072);   // 8 ints
  int* betti    = edgeCnt + NMAT;                                   // 8 ints

  sq_kernel   <<<dim3(4096),       dim3(256),  0, stream>>>(xb, xa, sq, edgeCnt);
  adj_kernel  <<<dim3(64, 256, 8), dim3(128),  0, stream>>>(xb, xa, sq, adj, edgeCnt);
  cc_kernel   <<<dim3(8),          dim3(1024), 0, stream>>>((const uint32_t*)adj, edgeCnt, betti);
  final_kernel<<<dim3(1),          dim3(32),   0, stream>>>(betti, (float*)d_out);
}